// GAT_72825465471459
// MI455X (gfx1250) — compile-verified
//
#include <hip/hip_runtime.h>
#include <math.h>

#define NN 100000
#define EE 1600000
#define ET (EE + NN)          // edges incl. self-loops = 1,700,000
#define IN_CH 128
#define HID 32
#define HEADS 4
#define HC 128                // HEADS*HID
#define OUT_CH 40
#define NEG_SLOPE 0.2f

typedef __attribute__((ext_vector_type(2))) float v2f;
typedef __attribute__((ext_vector_type(8))) float v8f;

__device__ __forceinline__ float atomic_max_f32(float* addr, float v) {
    if (v >= 0.0f)
        return __int_as_float(atomicMax((int*)addr, __float_as_int(v)));
    else
        return __uint_as_float(atomicMin((unsigned int*)addr, __float_as_uint(v)));
}

__device__ __forceinline__ float lrelu(float v) {
    return v >= 0.0f ? v : NEG_SLOPE * v;
}

// ---------------- GEMM 1: h1[N,128] = x[N,128] @ W1[128,128] ----------------
// 4 waves / block, one 16-row strip per wave, all 8 column tiles per wave.
__global__ void k_gemm1(const float* __restrict__ x, const float* __restrict__ W1,
                        float* __restrict__ h1) {
    const int wave = threadIdx.x >> 5;
    const int row0 = (blockIdx.x * 4 + wave) * 16;
    if (row0 >= NN) return;                 // wave-uniform
    const int lane = threadIdx.x & 31;
    const int m    = lane & 15;
    const int kb   = (lane >> 4) * 2;       // 0 or 2

    v8f c[8];
#pragma unroll
    for (int t = 0; t < 8; ++t) c[t] = (v8f){0,0,0,0,0,0,0,0};

    for (int k = 0; k < IN_CH; k += 4) {
        v2f a;
        const float* xr = x + (size_t)(row0 + m) * IN_CH + (k + kb);
        a.x = xr[0];
        a.y = xr[1];
#pragma unroll
        for (int t = 0; t < 8; ++t) {
            v2f b;
            const float* wr = W1 + (size_t)(k + kb) * HC + t * 16 + m;
            b.x = wr[0];
            b.y = wr[HC];
            c[t] = __builtin_amdgcn_wmma_f32_16x16x4_f32(false, a, false, b,
                                                         (short)0, c[t], false, false);
        }
    }
    const int rofs = (lane >> 4) * 8;
#pragma unroll
    for (int t = 0; t < 8; ++t)
#pragma unroll
        for (int r = 0; r < 8; ++r)
            h1[(size_t)(row0 + r + rofs) * HC + t * 16 + m] = c[t][r];
}

// ------------- per-node attention logits, layer 1: [N, HEADS] -------------
__global__ void k_al1(const float* __restrict__ h1, const float* __restrict__ a_src,
                      const float* __restrict__ a_dst, float* __restrict__ als,
                      float* __restrict__ ald) {
    int i = blockIdx.x * blockDim.x + threadIdx.x;
    if (i >= NN * HEADS) return;
    int n = i >> 2, h = i & 3;
    const float* hp = h1 + (size_t)n * HC + h * HID;
    float s = 0.f, d = 0.f;
#pragma unroll 8
    for (int cc = 0; cc < HID; ++cc) {
        float v = hp[cc];
        s += v * a_src[h * HID + cc];
        d += v * a_dst[h * HID + cc];
    }
    als[i] = s;
    ald[i] = d;
}

// ------------- init: agg1=0, m1=-inf, denom1=0 -------------
__global__ void k_init1(float* __restrict__ agg, float* __restrict__ m,
                        float* __restrict__ dn) {
    int i = blockIdx.x * blockDim.x + threadIdx.x;
    if (i < NN * HC) agg[i] = 0.0f;
    if (i < NN * HEADS) { m[i] = -__builtin_inff(); dn[i] = 0.0f; }
}

// ------------- edge pass: segment max, layer 1 -------------
__global__ void k_emax1(const int* __restrict__ ei, const float* __restrict__ als,
                        const float* __restrict__ ald, float* __restrict__ m) {
    int i = blockIdx.x * blockDim.x + threadIdx.x;
    if (i >= ET * HEADS) return;
    int e = i >> 2, h = i & 3;
    int s, d;
    if (e < EE) { s = ei[e]; d = ei[EE + e]; } else { s = d = e - EE; }
    float v = lrelu(als[s * 4 + h] + ald[d * 4 + h]);
    atomic_max_f32(&m[d * 4 + h], v);
}

// ------------- edge pass: exp-sum, layer 1 -------------
__global__ void k_esum1(const int* __restrict__ ei, const float* __restrict__ als,
                        const float* __restrict__ ald, const float* __restrict__ m,
                        float* __restrict__ dn) {
    int i = blockIdx.x * blockDim.x + threadIdx.x;
    if (i >= ET * HEADS) return;
    int e = i >> 2, h = i & 3;
    int s, d;
    if (e < EE) { s = ei[e]; d = ei[EE + e]; } else { s = d = e - EE; }
    float v = lrelu(als[s * 4 + h] + ald[d * 4 + h]);
    atomicAdd(&dn[d * 4 + h], __expf(v - m[d * 4 + h]));
}

// ------------- edge pass: weighted scatter-add, layer 1 (1 wave / edge) -------------
__global__ void k_eagg1(const int* __restrict__ ei, const float* __restrict__ h1,
                        const float* __restrict__ als, const float* __restrict__ ald,
                        const float* __restrict__ m, const float* __restrict__ dn,
                        float* __restrict__ agg) {
    int e = blockIdx.x * 8 + (threadIdx.x >> 5);
    if (e >= ET) return;                    // wave-uniform
    int lane = threadIdx.x & 31;
    int s, d;
    if (e < EE) { s = ei[e]; d = ei[EE + e]; } else { s = d = e - EE; }
    int h = lane >> 3;                      // 4 channels/lane, 8 lanes/head
    float v = lrelu(als[s * 4 + h] + ald[d * 4 + h]);
    float alpha = __expf(v - m[d * 4 + h]) / (dn[d * 4 + h] + 1e-16f);
    const float4 hv = *(const float4*)(h1 + (size_t)s * HC + lane * 4);
    float* op = agg + (size_t)d * HC + lane * 4;
    atomicAdd(op + 0, hv.x * alpha);
    atomicAdd(op + 1, hv.y * alpha);
    atomicAdd(op + 2, hv.z * alpha);
    atomicAdd(op + 3, hv.w * alpha);
}

// ------------- h2 = ELU(agg1 + b1)   (in-place) -------------
__global__ void k_fin1(float* __restrict__ agg, const float* __restrict__ b1) {
    int i = blockIdx.x * blockDim.x + threadIdx.x;
    if (i >= NN * HC) return;
    float v = agg[i] + b1[i & (HC - 1)];
    agg[i] = v > 0.0f ? v : expm1f(v);
}

// ---------------- GEMM 2: z[N,40] = h2[N,128] @ W2[128,40] ----------------
__global__ void k_gemm2(const float* __restrict__ h2, const float* __restrict__ W2,
                        float* __restrict__ z) {
    const int wave = threadIdx.x >> 5;
    const int row0 = (blockIdx.x * 4 + wave) * 16;
    if (row0 >= NN) return;                 // wave-uniform
    const int lane = threadIdx.x & 31;
    const int m    = lane & 15;
    const int kb   = (lane >> 4) * 2;

    v8f c[3];
#pragma unroll
    for (int t = 0; t < 3; ++t) c[t] = (v8f){0,0,0,0,0,0,0,0};

    for (int k = 0; k < HC; k += 4) {
        v2f a;
        const float* xr = h2 + (size_t)(row0 + m) * HC + (k + kb);
        a.x = xr[0];
        a.y = xr[1];
#pragma unroll
        for (int t = 0; t < 3; ++t) {
            int col = t * 16 + m;
            v2f b;
            b.x = (col < OUT_CH) ? W2[(size_t)(k + kb) * OUT_CH + col] : 0.0f;
            b.y = (col < OUT_CH) ? W2[(size_t)(k + kb + 1) * OUT_CH + col] : 0.0f;
            c[t] = __builtin_amdgcn_wmma_f32_16x16x4_f32(false, a, false, b,
                                                         (short)0, c[t], false, false);
        }
    }
    const int rofs = (lane >> 4) * 8;
#pragma unroll
    for (int t = 0; t < 3; ++t) {
        int col = t * 16 + m;
        if (col < OUT_CH)
#pragma unroll
            for (int r = 0; r < 8; ++r)
                z[(size_t)(row0 + r + rofs) * OUT_CH + col] = c[t][r];
    }
}

// ------------- per-node logits, layer 2 (1 head) -------------
__global__ void k_al2(const float* __restrict__ z, const float* __restrict__ a_src,
                      const float* __restrict__ a_dst, float* __restrict__ als,
                      float* __restrict__ ald) {
    int n = blockIdx.x * blockDim.x + threadIdx.x;
    if (n >= NN) return;
    const float* zp = z + (size_t)n * OUT_CH;
    float s = 0.f, d = 0.f;
#pragma unroll 8
    for (int cc = 0; cc < OUT_CH; ++cc) {
        float v = zp[cc];
        s += v * a_src[cc];
        d += v * a_dst[cc];
    }
    als[n] = s;
    ald[n] = d;
}

// ------------- init layer 2: out = b2 (broadcast), m2=-inf, denom2=0 -------------
__global__ void k_init2(float* __restrict__ out, const float* __restrict__ b2,
                        float* __restrict__ m, float* __restrict__ dn) {
    int i = blockIdx.x * blockDim.x + threadIdx.x;
    if (i < NN * OUT_CH) out[i] = b2[i % OUT_CH];
    if (i < NN) { m[i] = -__builtin_inff(); dn[i] = 0.0f; }
}

__global__ void k_emax2(const int* __restrict__ ei, const float* __restrict__ als,
                        const float* __restrict__ ald, float* __restrict__ m) {
    int e = blockIdx.x * blockDim.x + threadIdx.x;
    if (e >= ET) return;
    int s, d;
    if (e < EE) { s = ei[e]; d = ei[EE + e]; } else { s = d = e - EE; }
    atomic_max_f32(&m[d], lrelu(als[s] + ald[d]));
}

__global__ void k_esum2(const int* __restrict__ ei, const float* __restrict__ als,
                        const float* __restrict__ ald, const float* __restrict__ m,
                        float* __restrict__ dn) {
    int e = blockIdx.x * blockDim.x + threadIdx.x;
    if (e >= ET) return;
    int s, d;
    if (e < EE) { s = ei[e]; d = ei[EE + e]; } else { s = d = e - EE; }
    float v = lrelu(als[s] + ald[d]);
    atomicAdd(&dn[d], __expf(v - m[d]));
}

// 1 wave / edge; lanes cover channels {lane, lane+32(<40)}
__global__ void k_eagg2(const int* __restrict__ ei, const float* __restrict__ z,
                        const float* __restrict__ als, const float* __restrict__ ald,
                        const float* __restrict__ m, const float* __restrict__ dn,
                        float* __restrict__ out) {
    int e = blockIdx.x * 8 + (threadIdx.x >> 5);
    if (e >= ET) return;                    // wave-uniform
    int lane = threadIdx.x & 31;
    int s, d;
    if (e < EE) { s = ei[e]; d = ei[EE + e]; } else { s = d = e - EE; }
    float v = lrelu(als[s] + ald[d]);
    float alpha = __expf(v - m[d]) / (dn[d] + 1e-16f);
    const float* zp = z + (size_t)s * OUT_CH;
    float* op = out + (size_t)d * OUT_CH;
    atomicAdd(op + lane, zp[lane] * alpha);
    if (lane < OUT_CH - 32)
        atomicAdd(op + 32 + lane, zp[32 + lane] * alpha);
}

extern "C" void kernel_launch(void* const* d_in, const int* in_sizes, int n_in,
                              void* d_out, int out_size, void* d_ws, size_t ws_size,
                              hipStream_t stream) {
    const float* x      = (const float*)d_in[0];
    const int*   ei     = (const int*)  d_in[1];   // [2,E] row-major: [0..E)=src, [E..2E)=dst
    const float* W1     = (const float*)d_in[2];
    const float* a_src1 = (const float*)d_in[3];
    const float* a_dst1 = (const float*)d_in[4];
    const float* b1     = (const float*)d_in[5];
    const float* W2     = (const float*)d_in[6];
    const float* a_src2 = (const float*)d_in[7];
    const float* a_dst2 = (const float*)d_in[8];
    const float* b2     = (const float*)d_in[9];
    float* out = (float*)d_out;

    // workspace carve-up (floats)
    float* ws   = (float*)d_ws;
    float* h1   = ws;                          // N*HC
    float* agg1 = h1   + (size_t)NN * HC;      // N*HC (becomes h2 after ELU)
    float* z2   = agg1 + (size_t)NN * HC;      // N*OUT_CH
    float* als1 = z2   + (size_t)NN * OUT_CH;  // N*HEADS
    float* ald1 = als1 + (size_t)NN * HEADS;
    float* m1   = ald1 + (size_t)NN * HEADS;
    float* dn1  = m1   + (size_t)NN * HEADS;
    float* als2 = dn1  + (size_t)NN * HEADS;   // N
    float* ald2 = als2 + NN;
    float* m2   = ald2 + NN;
    float* dn2  = m2   + NN;

    const int TB = 256;
    const int gemmBlocks = (NN / 16 + 3) / 4;          // 4 waves/block, 16 rows/wave
    const int edgeWaveBlocks = (ET + 7) / 8;           // 8 waves/block, 1 edge/wave

    // ---- layer 1 ----
    k_gemm1<<<gemmBlocks, 128, 0, stream>>>(x, W1, h1);
    k_al1  <<<(NN * HEADS + TB - 1) / TB, TB, 0, stream>>>(h1, a_src1, a_dst1, als1, ald1);
    k_init1<<<(NN * HC + TB - 1) / TB, TB, 0, stream>>>(agg1, m1, dn1);
    k_emax1<<<(ET * HEADS + TB - 1) / TB, TB, 0, stream>>>(ei, als1, ald1, m1);
    k_esum1<<<(ET * HEADS + TB - 1) / TB, TB, 0, stream>>>(ei, als1, ald1, m1, dn1);
    k_eagg1<<<edgeWaveBlocks, 256, 0, stream>>>(ei, h1, als1, ald1, m1, dn1, agg1);
    k_fin1 <<<(NN * HC + TB - 1) / TB, TB, 0, stream>>>(agg1, b1);   // agg1 -> h2

    // ---- layer 2 ----
    k_gemm2<<<gemmBlocks, 128, 0, stream>>>(agg1, W2, z2);
    k_al2  <<<(NN + TB - 1) / TB, TB, 0, stream>>>(z2, a_src2, a_dst2, als2, ald2);
    k_init2<<<(NN * OUT_CH + TB - 1) / TB, TB, 0, stream>>>(out, b2, m2, dn2);
    k_emax2<<<(ET + TB - 1) / TB, TB, 0, stream>>>(ei, als2, ald2, m2);
    k_esum2<<<(ET + TB - 1) / TB, TB, 0, stream>>>(ei, als2, ald2, m2, dn2);
    k_eagg2<<<edgeWaveBlocks, 256, 0, stream>>>(ei, z2, als2, ald2, m2, dn2, out);
}